// TinyRGATLayer_30614526885989
// MI455X (gfx1250) — compile-verified
//
#include <hip/hip_runtime.h>
#include <math.h>

#define HID 128
#define NUM_RELS 64
#define REL_DIM 16

typedef float v2f __attribute__((ext_vector_type(2)));
typedef float v8f __attribute__((ext_vector_type(8)));

// ---- order-preserving float <-> uint key (for atomic max on floats) ----
__device__ __forceinline__ unsigned fkey(float f) {
    unsigned u = __float_as_uint(f);
    return (u & 0x80000000u) ? ~u : (u | 0x80000000u);
}
__device__ __forceinline__ float fkeyinv(unsigned k) {
    unsigned u = (k & 0x80000000u) ? (k & 0x7fffffffu) : ~k;
    return __uint_as_float(u);
}

// ---- init: out[n][c] = bias[c] ----
__global__ void k_init_out(float* __restrict__ out, const float* __restrict__ bias, int total) {
    int i = blockIdx.x * blockDim.x + threadIdx.x;
    if (i < total) out[i] = bias[i & (HID - 1)];
}

// ---- init: maxkey = 0 (below key(-inf)), denom = 0 ----
__global__ void k_init_scalars(unsigned* __restrict__ maxkey, float* __restrict__ denom, int N) {
    int i = blockIdx.x * blockDim.x + threadIdx.x;
    if (i < N) { maxkey[i] = 0u; denom[i] = 0.0f; }
}

// ---- sr[t] = rel_emb[t] . (W_rel^T @ a_r)   (single 64-thread block) ----
__global__ void k_rel(const float* __restrict__ rel, const float* __restrict__ Wr,
                      const float* __restrict__ att, float* __restrict__ sr) {
    __shared__ float v[REL_DIM];
    int t = threadIdx.x;
    if (t < REL_DIM) {
        const float* ar = att + 2 * HID;
        float acc = 0.0f;
        for (int h = 0; h < HID; ++h) acc += ar[h] * Wr[h * REL_DIM + t];
        v[t] = acc;
    }
    __syncthreads();
    if (t < NUM_RELS) {
        float acc = 0.0f;
        #pragma unroll
        for (int d = 0; d < REL_DIM; ++d) acc += rel[t * REL_DIM + d] * v[d];
        sr[t] = acc;
    }
}

// ---- h = x @ W_msg^T via V_WMMA_F32_16X16X4_F32 ----
// Block = 256 threads = 8 waves, computes 64 rows x 128 cols.
// The 64x128 A-tile is shared by all 8 waves -> staged once in LDS (32 KB).
// Each lane's B data (64 floats of one W_msg row) is preloaded into 64 VGPRs
// via 32 immediate-offset global_load_b64, so the K-loop is ds_load + wmma only.
__global__ void __launch_bounds__(256) k_gemm(const float* __restrict__ x,
                                              const float* __restrict__ Wm,
                                              float* __restrict__ h, int nrows) {
    __shared__ float xs[64 * HID];            // 32 KB of the 320 KB/WGP LDS

    const int tid  = threadIdx.x;
    const int lane = tid & 31;
    const int wave = tid >> 5;
    const int row0 = blockIdx.x * 64;
    const int col0 = wave * 16;
    const int lrow = lane & 15;               // A: M index / B: N index
    const int koff = (lane >> 4) * 2;         // lanes 16-31 hold K = {2,3}

    // ---- preload B fragments: breg[s] covers K = 4s+koff .. 4s+koff+1 ----
    const float* bptr = Wm + (size_t)(col0 + lrow) * HID + koff;
    v2f breg[32];
    #pragma unroll
    for (int s = 0; s < 32; ++s) breg[s] = *(const v2f*)(bptr + 4 * s);

    // ---- stage A-tile (64 rows of x) into LDS, coalesced float4 copy ----
    const int rows_here = (nrows - row0 >= 64) ? 64 : (nrows - row0);
    const int validf4   = rows_here * (HID / 4);          // valid float4s in tile
    const float4* src4  = (const float4*)(x + (size_t)row0 * HID);
    #pragma unroll
    for (int i = 0; i < 8; ++i) {
        int idx = i * 256 + tid;                          // 2048 float4s total
        int s   = (idx < validf4) ? idx : (validf4 - 1);  // clamp tail reads
        ((float4*)xs)[idx] = src4[s];
    }
    __syncthreads();

    // ---- K-loop: ds_load_b64 (immediate offsets) + wmma, fully unrolled ----
    const float* ap = xs + lrow * HID + koff;             // per-lane LDS base
    v8f acc[4] = {v8f{}, v8f{}, v8f{}, v8f{}};
    #pragma unroll
    for (int s = 0; s < 32; ++s) {
        #pragma unroll
        for (int mi = 0; mi < 4; ++mi) {
            v2f a = *(const v2f*)(ap + mi * 16 * HID + 4 * s);   // A(m, 4s+koff..+1)
            acc[mi] = __builtin_amdgcn_wmma_f32_16x16x4_f32(
                false, a, false, breg[s], (short)0, acc[mi], false, false);
        }
    }

    // C/D layout: lane holds col n=lane&15; VGPR v holds row v + 8*(lane>>4)
    const int n  = lane & 15;
    const int mb = (lane >> 4) * 8;
    #pragma unroll
    for (int mi = 0; mi < 4; ++mi) {
        #pragma unroll
        for (int vv = 0; vv < 8; ++vv) {
            int rr = row0 + 16 * mi + mb + vv;
            if (rr < nrows) h[(size_t)rr * HID + col0 + n] = acc[mi][vv];
        }
    }
}

// ---- sd[n] = h[n].a_d, ss[n] = h[n].a_s : one wave32 per node ----
__global__ void k_nodescores(const float* __restrict__ h, const float* __restrict__ att,
                             float* __restrict__ sd, float* __restrict__ ss, int N) {
    int gid  = blockIdx.x * blockDim.x + threadIdx.x;
    int node = gid >> 5;
    int lane = gid & 31;
    if (node >= N) return;
    float4 v  = ((const float4*)(h + (size_t)node * HID))[lane];
    float4 ad = ((const float4*)(att))[lane];
    float4 as = ((const float4*)(att + HID))[lane];
    float pd = v.x * ad.x + v.y * ad.y + v.z * ad.z + v.w * ad.w;
    float ps = v.x * as.x + v.y * as.y + v.z * as.z + v.w * as.w;
    #pragma unroll
    for (int off = 16; off > 0; off >>= 1) {
        pd += __shfl_down(pd, off, 32);
        ps += __shfl_down(ps, off, 32);
    }
    if (lane == 0) { sd[node] = pd; ss[node] = ps; }
}

// ---- pass 1 over edges: raw score + segmented atomic max ----
__global__ void k_edge_max(const long long* __restrict__ ei, const long long* __restrict__ et,
                           const float* __restrict__ ea,
                           const float* __restrict__ sd, const float* __restrict__ ss,
                           const float* __restrict__ sr,
                           float* __restrict__ e_raw, unsigned* __restrict__ maxkey, int E) {
    int i = blockIdx.x * blockDim.x + threadIdx.x;
    if (i >= E) return;
    long long s = ei[i];
    long long d = ei[(size_t)E + i];
    long long t = et[i];
    t = (t < 0) ? 0 : ((t > NUM_RELS - 1) ? NUM_RELS - 1 : t);
    float e = sd[d] + ss[s] + sr[t];
    e = (e > 0.0f) ? e : 0.2f * e;                 // leaky_relu(0.2)
    float conf = fmaxf(ea[i], 1e-6f);
    e += 0.5f * __logf(conf);
    e_raw[i] = e;
    atomicMax(maxkey + d, fkey(e));
}

// ---- pass 2 over edges: exp(e - max[dst]) + segmented atomic sum ----
__global__ void k_edge_exp(const long long* __restrict__ ei, const float* __restrict__ e_raw,
                           const unsigned* __restrict__ maxkey,
                           float* __restrict__ e_exp, float* __restrict__ denom, int E) {
    int i = blockIdx.x * blockDim.x + threadIdx.x;
    if (i >= E) return;
    long long d = ei[(size_t)E + i];
    float m  = fkeyinv(maxkey[d]);
    float ee = __expf(e_raw[i] - m);
    e_exp[i] = ee;
    atomicAdd(denom + d, ee);
}

// ---- pass 3: out[dst] += alpha * h[src] : one wave32 per edge, float4/lane ----
__global__ void k_scatter(const float* __restrict__ h, const float* __restrict__ e_exp,
                          const float* __restrict__ denom, const long long* __restrict__ ei,
                          float* __restrict__ out, int E) {
    int gid  = blockIdx.x * blockDim.x + threadIdx.x;
    int e    = gid >> 5;
    int lane = gid & 31;
    if (e >= E) return;
    long long s = ei[e];
    long long d = ei[(size_t)E + e];
    float alpha = e_exp[e] / denom[d];
    float4 v = ((const float4*)(h + (size_t)s * HID))[lane];
    float* op = out + (size_t)d * HID + lane * 4;
    atomicAdd(op + 0, alpha * v.x);
    atomicAdd(op + 1, alpha * v.y);
    atomicAdd(op + 2, alpha * v.z);
    atomicAdd(op + 3, alpha * v.w);
}

static inline size_t align256(size_t v) { return (v + 255) & ~(size_t)255; }

extern "C" void kernel_launch(void* const* d_in, const int* in_sizes, int n_in,
                              void* d_out, int out_size, void* d_ws, size_t ws_size,
                              hipStream_t stream) {
    const float*     x    = (const float*)d_in[0];      // [N,128]
    const float*     Wm   = (const float*)d_in[1];      // [128,128]
    const float*     rel  = (const float*)d_in[2];      // [64,16]
    const float*     Wr   = (const float*)d_in[3];      // [128,16]
    const float*     att  = (const float*)d_in[4];      // [384]
    const float*     bias = (const float*)d_in[5];      // [128]
    const float*     ea   = (const float*)d_in[6];      // [E,1]
    const long long* ei   = (const long long*)d_in[7];  // [2,E] int64
    const long long* et   = (const long long*)d_in[8];  // [E]   int64

    const int N = in_sizes[0] / HID;
    const int E = in_sizes[8];
    float* out = (float*)d_out;

    // workspace carve-up (~31 MB total)
    char*  w   = (char*)d_ws;
    size_t off = 0;
    float*    h      = (float*)(w + off);  off = align256(off + (size_t)N * HID * 4);
    float*    sd     = (float*)(w + off);  off = align256(off + (size_t)N * 4);
    float*    ss     = (float*)(w + off);  off = align256(off + (size_t)N * 4);
    float*    sr     = (float*)(w + off);  off = align256(off + (size_t)NUM_RELS * 4);
    float*    e_raw  = (float*)(w + off);  off = align256(off + (size_t)E * 4);
    float*    e_exp  = (float*)(w + off);  off = align256(off + (size_t)E * 4);
    unsigned* maxkey = (unsigned*)(w + off); off = align256(off + (size_t)N * 4);
    float*    denom  = (float*)(w + off);  off = align256(off + (size_t)N * 4);
    (void)ws_size; (void)n_in; (void)out_size;

    const int B = 256;
    // init
    k_init_out<<<((size_t)N * HID + B - 1) / B, B, 0, stream>>>(out, bias, N * HID);
    k_init_scalars<<<(N + B - 1) / B, B, 0, stream>>>(maxkey, denom, N);
    // tiny relation-score precompute
    k_rel<<<1, 64, 0, stream>>>(rel, Wr, att, sr);
    // node projection (WMMA, LDS-staged A, register-resident B)
    k_gemm<<<(N + 63) / 64, 256, 0, stream>>>(x, Wm, h, N);
    // per-node attention scalars
    k_nodescores<<<((size_t)N * 32 + B - 1) / B, B, 0, stream>>>(h, att, sd, ss, N);
    // segmented softmax over edges
    k_edge_max<<<(E + B - 1) / B, B, 0, stream>>>(ei, et, ea, sd, ss, sr, e_raw, maxkey, E);
    k_edge_exp<<<(E + B - 1) / B, B, 0, stream>>>(ei, e_raw, maxkey, e_exp, denom, E);
    // weighted message scatter
    k_scatter<<<((size_t)E * 32 + B - 1) / B, B, 0, stream>>>(h, e_exp, denom, ei, out, E);
}